// HDHOGLayer_1297080123474
// MI455X (gfx1250) — compile-verified
//
#include <hip/hip_runtime.h>
#include <hip/hip_bf16.h>
#include <math.h>

#ifndef __has_builtin
#define __has_builtin(x) 0
#endif

// ---------------- problem geometry ----------------
#define IMG_W 4096
#define IMG_H 4096
#define NCX 512          // cells per row   (IMG_W / 8)
#define NCY 512          // cell rows       (IMG_H / 8)
#define NB  9            // orientation bins
#define NBR 510          // block rows  (NCY - 3 + 1)
#define NBC 510          // block cols  (NCX - 3 + 1)
#define ROW_OUT (NBC * 81)   // 41310 output floats per block-row

// ---------------- kernel-1 tiling -----------------
#define TCOLS   128      // image columns per workgroup (16 cells)
#define LROWS   10       // 8 rows + 1 halo row above/below
#define LCOLS   136      // 128 cols + 4 halo cols each side (16B aligned)
#define LCHUNKS (LCOLS / 4)           // b128 chunks per LDS row
#define NCHUNK  (LROWS * LCHUNKS)     // total b128 chunks per tile

// vector-of-4-int type matching the builtin's expected pointee
typedef int v4i_ __attribute__((vector_size(16)));
typedef __attribute__((address_space(1))) v4i_* gv4i_p;
typedef __attribute__((address_space(3))) v4i_* lv4i_p;

// ---- CDNA5 async global->LDS copy (ASYNCcnt path) ----
__device__ __forceinline__ void async_copy_b128(const float* gsrc, float* lds_dst) {
#if __has_builtin(__builtin_amdgcn_global_load_async_to_lds_b128)
    __builtin_amdgcn_global_load_async_to_lds_b128(
        (gv4i_p)gsrc, (lv4i_p)lds_dst, /*imm offset*/0, /*cpol*/0);
#else
    unsigned laddr = (unsigned)(unsigned long long)lds_dst;   // low 32 bits = LDS addr
    asm volatile("global_load_async_to_lds_b128 %0, %1, off"
                 :: "v"(laddr), "v"(gsrc)
                 : "memory");
#endif
}

__device__ __forceinline__ void wait_async_zero() {
#if __has_builtin(__builtin_amdgcn_s_wait_asynccnt)
    __builtin_amdgcn_s_wait_asynccnt(0);
#else
    asm volatile("s_wait_asynccnt 0" ::: "memory");
#endif
}

// =====================================================================
// Kernel 1: per-cell 9-bin orientation histograms (means over 8x8 cell)
//   grid = (IMG_W / TCOLS, NCY), block = 128 threads (4 waves, wave32)
//   hist layout: hist[cy][cx][bin], bin innermost
// =====================================================================
__global__ __launch_bounds__(TCOLS)
void hog_cell_hist(const float* __restrict__ x, float* __restrict__ hist) {
    __shared__ float tile[LROWS * LCOLS];

    const int tid = threadIdx.x;               // 0..127
    const int c0  = blockIdx.x * TCOLS;        // first image column of tile
    const int cy  = blockIdx.y;                // cell row
    const int r0  = cy * 8;                    // first image row of cell strip

    // ---- stage 10 x 136 floats into LDS with async global->LDS copies ----
    for (int i = tid; i < NCHUNK; i += TCOLS) {
        const int lr  = i / LCHUNKS;           // 0..9  (row r0-1+lr)
        const int lc4 = i % LCHUNKS;           // chunk in row
        int gr = r0 - 1 + lr;
        gr = gr < 0 ? 0 : (gr > IMG_H - 1 ? IMG_H - 1 : gr);
        int gc = c0 - 4 + lc4 * 4;
        gc = gc < 0 ? 0 : (gc > IMG_W - 4 ? IMG_W - 4 : gc);
        // clamped halo values are never consumed (border gradients are masked)
        async_copy_b128(x + (size_t)gr * IMG_W + gc, &tile[lr * LCOLS + lc4 * 4]);
    }
    wait_async_zero();
    __syncthreads();

    // ---- per-lane: one column, 8 rows; 9 VGPR bin accumulators ----
    const int ct = tid;                        // column within tile (0..127)
    const int gc = c0 + ct;                    // global column
    const int lc = ct + 4;                     // LDS column

    float acc[NB];
#pragma unroll
    for (int b = 0; b < NB; ++b) acc[b] = 0.0f;

#pragma unroll
    for (int rr = 0; rr < 8; ++rr) {
        const int gr = r0 + rr;
        const int lr = rr + 1;
        const float up = sqrtf(tile[(lr - 1) * LCOLS + lc]);
        const float dn = sqrtf(tile[(lr + 1) * LCOLS + lc]);
        const float lf = sqrtf(tile[lr * LCOLS + lc - 1]);
        const float rt = sqrtf(tile[lr * LCOLS + lc + 1]);

        const float grow = (gr == 0 || gr == IMG_H - 1) ? 0.0f : (dn - up);
        const float gcol = (gc == 0 || gc == IMG_W - 1) ? 0.0f : (rt - lf);

        const float mag = sqrtf(grow * grow + gcol * gcol);
        float deg = atan2f(grow, gcol) * 57.29577951308232f;     // rad2deg
        deg = deg - 180.0f * floorf(deg * (1.0f / 180.0f));      // python mod -> [0,180)
        int bin = (int)(deg * (1.0f / 20.0f));
        bin = bin > 8 ? 8 : (bin < 0 ? 0 : bin);

#pragma unroll
        for (int b = 0; b < NB; ++b)
            acc[b] += (bin == b) ? mag : 0.0f;
    }

    // ---- reduce 8 columns of a cell across lanes (wave32 ds-permute) ----
#pragma unroll
    for (int b = 0; b < NB; ++b) {
        float v = acc[b];
        v += __shfl_xor(v, 1, 32);
        v += __shfl_xor(v, 2, 32);
        v += __shfl_xor(v, 4, 32);
        acc[b] = v;
    }

    if ((tid & 7) == 0) {
        const int cellx = (c0 >> 3) + (ct >> 3);
        float* hp = hist + ((size_t)cy * NCX + cellx) * NB;
#pragma unroll
        for (int b = 0; b < NB; ++b)
            hp[b] = acc[b] * (1.0f / 64.0f);      // mean over 8x8 cell
    }
}

// =====================================================================
// Kernel 2: 3x3-cell block L2 normalization, store-coalesced
//   one workgroup per block-row r (510 blocks x 256 threads)
//   phase 1: per-column inverse norms into LDS (hist is L2-resident)
//   phase 2: linear sweep of the row's 41310 outputs -> every store
//            instruction writes 32 contiguous floats (one 128B line)
// =====================================================================
__global__ __launch_bounds__(256)
void hog_block_norm(const float* __restrict__ hist, float* __restrict__ out) {
    __shared__ float scale[NBC];

    const int r = blockIdx.x;                  // 0..509

    // ---- phase 1: scale[c] = 1/sqrt(ssq + EPS^2) ----
    for (int c = threadIdx.x; c < NBC; c += 256) {
        const float* hb = hist + ((size_t)r * NCX + c) * NB;
        float ssq = 0.0f;
#pragma unroll
        for (int i = 0; i < 3; ++i)
#pragma unroll
            for (int j = 0; j < 3; ++j)
#pragma unroll
                for (int o = 0; o < NB; ++o) {
                    const float t = hb[((size_t)i * NCX + j) * NB + o];
                    ssq += t * t;
                }
        scale[c] = 1.0f / sqrtf(ssq + 1e-10f);   // EPS^2 = 1e-10
    }
    __syncthreads();

    // ---- phase 2: coalesced linear output sweep ----
    float* ob = out + (size_t)r * ROW_OUT;
    for (int p = threadIdx.x; p < ROW_OUT; p += 256) {
        const int c  = p / 81;                 // block column
        const int k  = p - c * 81;             // 0..80 within block
        const int i  = k / 27;                 // cell row in block
        const int jo = k - i * 27;
        const int j  = jo / 9;                 // cell col in block
        const int o  = jo - j * 9;             // orientation bin
        const float t = hist[((size_t)(r + i) * NCX + (c + j)) * NB + o];
        ob[p] = t * scale[c];
    }
}

// =====================================================================
extern "C" void kernel_launch(void* const* d_in, const int* in_sizes, int n_in,
                              void* d_out, int out_size, void* d_ws, size_t ws_size,
                              hipStream_t stream) {
    (void)in_sizes; (void)n_in; (void)out_size; (void)ws_size;

    const float* x    = (const float*)d_in[0];
    float*       out  = (float*)d_out;
    float*       hist = (float*)d_ws;   // 512*512*9 f32 = 9.4 MB scratch

    dim3 g1(IMG_W / TCOLS, NCY);        // (32, 512)
    hog_cell_hist<<<g1, TCOLS, 0, stream>>>(x, hist);

    hog_block_norm<<<NBR, 256, 0, stream>>>(hist, out);
}